// CapsuleLayer_25288767439054
// MI455X (gfx1250) — compile-verified
//
#include <hip/hip_runtime.h>

// CDNA5 / gfx1250 fused capsule-routing kernel (round 2).
// priors are never materialized: recomputed from L2-resident x (18.9MB) /
// W (23.6MB) with V_WMMA_F32_16X16X4_F32 in *transposed* orientation
// (D'[o,b] = W^T x), which makes the ||priors-out||^2 reduction in-lane.
// Routing softmax weights live as E=exp(L) in the 320KB WGP LDS.

typedef __attribute__((ext_vector_type(2))) float v2f;
typedef __attribute__((ext_vector_type(8))) float v8f;

#define WMMA(a, b, c) \
  __builtin_amdgcn_wmma_f32_16x16x4_f32(false, (a), false, (b), (short)0, (c), false, false)

constexpr int Bsz = 128, Csz = 10, Rsz = 4608, Isz = 8, Osz = 16;
constexpr int THREADS = 512, WAVES = 16, RPW = Rsz / WAVES;  // 288 r per wave

// E row stride padded: 4610 mod 64 = 2 -> column accesses hit distinct banks.
constexpr int RSTRIDE  = Rsz + 2;                 // 4610
constexpr int LDS_E    = 16 * RSTRIDE;            // 73760 floats
constexpr int LDS_SRED = 4096;                    // per-wave s' tiles / partials
constexpr int LDS_TOTAL = LDS_E + LDS_SRED + 16 + 256;  // 78128 f = 312512 B

__global__ __launch_bounds__(THREADS)
void caps_route_kernel(const float* __restrict__ x,     // [128,4608,8]
                       const float* __restrict__ w,     // [10,4608,8,16]
                       const float* __restrict__ bias,  // [10,16]
                       float* __restrict__ out)         // [128,10,16]
{
  extern __shared__ float smem[];
  float* El   = smem;                  // E[b][r] = exp(routing logit)
  float* sred = smem + LDS_E;
  float* Sinv = sred + LDS_SRED;
  float* outl = Sinv + 16;             // outputs, transposed: outl[o*16+b]

  const int btile = blockIdx.x;        // 0..7
  const int c     = blockIdx.y;        // 0..9
  const int tid   = threadIdx.x;
  const int lane  = tid & 31;
  const int wv    = tid >> 5;
  const int half  = lane >> 4;         // K-half for frags / o-half for D'
  const int col   = lane & 15;         // A-row (o) == B/D' col (b)
  const int bglob = btile * 16 + col;

  for (int idx = tid; idx < LDS_E; idx += THREADS) El[idx] = 1.0f;  // exp(0)
  __syncthreads();

  const int r0 = wv * RPW, r1 = r0 + RPW;
  const float* xb = x + (size_t)bglob * Rsz * Isz;    // x[bglob, r, i]
  const float* wb = w + (size_t)c * Rsz * Isz * Osz;  // w[c, r, i, n]

  v8f out_frag;                        // outputs in D' layout: [o=j+8h][b=col]
  #pragma unroll
  for (int j = 0; j < 8; ++j) out_frag[j] = 0.f;

  for (int it = 0; it < 3; ++it) {
    // ------- d2 pass: E[b,r] *= exp(exp(-||priors[b,r,:]-out[b,:]||^2)) ----
    if (it > 0) {
      for (int r = r0; r < r1; r += 2) {
        const float* xr0 = xb + (size_t)r * Isz;
        const float* xr1 = xr0 + Isz;
        const float* wr0 = wb + (size_t)r * Isz * Osz;
        const float* wr1 = wr0 + Isz * Osz;
        // r : A = W^T slice (m=o,k=i), B = x^T (k=i,n=b)
        v2f wa0, wa1, xa0, xa1;
        wa0.x = wr0[(2 * half + 0) * Osz + col];
        wa0.y = wr0[(2 * half + 1) * Osz + col];
        wa1.x = wr0[(4 + 2 * half) * Osz + col];
        wa1.y = wr0[(5 + 2 * half) * Osz + col];
        xa0 = *(const v2f*)(xr0 + 2 * half);
        xa1 = *(const v2f*)(xr0 + 4 + 2 * half);
        // r+1
        v2f wb0, wb1, xb0, xb1;
        wb0.x = wr1[(2 * half + 0) * Osz + col];
        wb0.y = wr1[(2 * half + 1) * Osz + col];
        wb1.x = wr1[(4 + 2 * half) * Osz + col];
        wb1.y = wr1[(5 + 2 * half) * Osz + col];
        xb0 = *(const v2f*)(xr1 + 2 * half);
        xb1 = *(const v2f*)(xr1 + 4 + 2 * half);

        v8f d0, d1;                          // two independent WMMA chains
        #pragma unroll
        for (int j = 0; j < 8; ++j) { d0[j] = 0.f; d1[j] = 0.f; }
        d0 = WMMA(wa0, xa0, d0);  d1 = WMMA(wb0, xb0, d1);
        d0 = WMMA(wa1, xa1, d0);  d1 = WMMA(wb1, xb1, d1);

        float u0 = 0.f, u1 = 0.f;            // in-lane partial over 8 o's
        #pragma unroll
        for (int j = 0; j < 8; ++j) {
          float t0 = d0[j] - out_frag[j];  u0 = fmaf(t0, t0, u0);
          float t1 = d1[j] - out_frag[j];  u1 = fmaf(t1, t1, u1);
        }
        u0 += __shfl_xor(u0, 16);            // merge o-halves
        u1 += __shfl_xor(u1, 16);
        float u    = half ? u1 : u0;         // lanes 0-15: r, 16-31: r+1
        float mult = __expf(__expf(-u));     // sigma = 1
        int addr = col * RSTRIDE + r + half; // 32 distinct banks
        El[addr] *= mult;
      }
      __syncthreads();
    }

    // ------- softmax denominator: Sinv[b] = 1 / sum_r E[b,r] --------------
    {
      int b = tid >> 5, sub = tid & 31;
      float acc = 0.f;
      for (int r = sub; r < Rsz; r += 32) acc += El[b * RSTRIDE + r];
      sred[tid] = acc;
      __syncthreads();
      if (tid < 16) {
        float s = 0.f;
        for (int k = 0; k < 32; ++k) s += sred[tid * 32 + k];
        Sinv[tid] = 1.0f / s;
      }
      __syncthreads();
    }

    // ------- s pass: s'[o,b] = sum_{r,k} W[r,k,o] * (prob*x)[b,r,k] -------
    {
      float sinv = Sinv[col];
      v8f a00, a01, a10, a11;                // 4 independent accumulators
      #pragma unroll
      for (int j = 0; j < 8; ++j) { a00[j]=0.f; a01[j]=0.f; a10[j]=0.f; a11[j]=0.f; }
      for (int r = r0; r < r1; r += 2) {
        float p0 = El[col * RSTRIDE + r]     * sinv;
        float p1 = El[col * RSTRIDE + r + 1] * sinv;
        const float* xr0 = xb + (size_t)r * Isz;
        const float* xr1 = xr0 + Isz;
        const float* wr0 = wb + (size_t)r * Isz * Osz;
        const float* wr1 = wr0 + Isz * Osz;
        v2f wa0, wa1, xa0, xa1, wb0, wb1, xb0, xb1;
        wa0.x = wr0[(2 * half + 0) * Osz + col];
        wa0.y = wr0[(2 * half + 1) * Osz + col];
        wa1.x = wr0[(4 + 2 * half) * Osz + col];
        wa1.y = wr0[(5 + 2 * half) * Osz + col];
        xa0 = *(const v2f*)(xr0 + 2 * half);
        xa1 = *(const v2f*)(xr0 + 4 + 2 * half);
        wb0.x = wr1[(2 * half + 0) * Osz + col];
        wb0.y = wr1[(2 * half + 1) * Osz + col];
        wb1.x = wr1[(4 + 2 * half) * Osz + col];
        wb1.y = wr1[(5 + 2 * half) * Osz + col];
        xb0 = *(const v2f*)(xr1 + 2 * half);
        xb1 = *(const v2f*)(xr1 + 4 + 2 * half);
        xa0.x *= p0; xa0.y *= p0; xa1.x *= p0; xa1.y *= p0;  // fold probs into B
        xb0.x *= p1; xb0.y *= p1; xb1.x *= p1; xb1.y *= p1;
        a00 = WMMA(wa0, xa0, a00);  a01 = WMMA(wa1, xa1, a01);
        a10 = WMMA(wb0, xb0, a10);  a11 = WMMA(wb1, xb1, a11);
      }
      v8f sacc = (a00 + a01) + (a10 + a11);
      #pragma unroll
      for (int j = 0; j < 8; ++j)
        sred[wv * 256 + (j + 8 * half) * 16 + col] = sacc[j];
      __syncthreads();
      if (tid < 256) {                        // cross-wave reduce, t=(o*16+b)
        float s = 0.f;
        for (int k = 0; k < WAVES; ++k) s += sred[k * 256 + tid];
        sred[tid] = s;
      }
      __syncthreads();
      if (tid < 16) {                         // squash + bias, b = tid
        float sq = 0.f;
        for (int o = 0; o < 16; ++o) { float v = sred[o * 16 + tid]; sq += v * v; }
        float scale = (sq / (1.f + sq)) * rsqrtf(sq);
        for (int o = 0; o < 16; ++o)
          outl[o * 16 + tid] = sred[o * 16 + tid] * scale + bias[c * 16 + o];
      }
      __syncthreads();
      #pragma unroll
      for (int j = 0; j < 8; ++j)             // outputs tile in D' layout
        out_frag[j] = outl[(j + 8 * half) * 16 + col];
    }

    if (it == 2 && tid < 256) {               // final store [B,C,O], coalesced
      int b = tid >> 4, o = tid & 15;
      out[((size_t)(btile * 16 + b) * Csz + c) * Osz + o] = outl[o * 16 + b];
    }
  }
}

extern "C" void kernel_launch(void* const* d_in, const int* in_sizes, int n_in,
                              void* d_out, int out_size, void* d_ws, size_t ws_size,
                              hipStream_t stream) {
  (void)in_sizes; (void)n_in; (void)out_size; (void)d_ws; (void)ws_size;
  const float* x    = (const float*)d_in[0];
  const float* w    = (const float*)d_in[1];
  const float* bias = (const float*)d_in[2];
  float* out        = (float*)d_out;

  dim3 grid(Bsz / 16, Csz);                   // 8 x 10 = 80 workgroups
  size_t lds_bytes = (size_t)LDS_TOTAL * sizeof(float);  // 312,512 B < 320 KB
  caps_route_kernel<<<grid, THREADS, lds_bytes, stream>>>(x, w, bias, out);
}